// LFGNN_14894946583442
// MI455X (gfx1250) — compile-verified
//
#include <hip/hip_runtime.h>
#include <hip/hip_bf16.h>

typedef __attribute__((ext_vector_type(2))) float v2f;
typedef __attribute__((ext_vector_type(8))) float v8f;

// ---------------------------------------------------------------------------
// Zero-fill (grid-stride)
// ---------------------------------------------------------------------------
__global__ void zero_kernel(float* __restrict__ p, long n) {
    long i = (long)blockIdx.x * blockDim.x + threadIdx.x;
    long stride = (long)gridDim.x * blockDim.x;
    for (; i < n; i += stride) p[i] = 0.0f;
}

// ---------------------------------------------------------------------------
// Scatter-add of 64-wide feature rows: agg[dst] += feat[src] over all edges.
// thread -> (edge, feature); contiguous lanes hit contiguous features so both
// the gather loads and the float atomics are lane-coalesced. Everything stays
// L2-resident (feat + agg = 51 MB << 192 MB L2).
// ---------------------------------------------------------------------------
__global__ void scatter_add64(const float* __restrict__ feat,
                              const int* __restrict__ src,
                              const int* __restrict__ dst,
                              float* __restrict__ agg, long nWork) {
    long tid = (long)blockIdx.x * blockDim.x + threadIdx.x;
    if (tid >= nWork) return;
    long e = tid >> 6;
    int  f = (int)(tid & 63);
    long s = src[e];
    long d = dst[e];
    atomicAdd(&agg[d * 64 + f], feat[s * 64 + f]);
}

// Scatter-add of 2-wide rows (layer 3, pre-projected features)
__global__ void scatter_add2(const float* __restrict__ feat,
                             const int* __restrict__ src,
                             const int* __restrict__ dst,
                             float* __restrict__ out, long nWork) {
    long tid = (long)blockIdx.x * blockDim.x + threadIdx.x;
    if (tid >= nWork) return;
    long e = tid >> 1;
    int  c = (int)(tid & 1);
    long s = src[e];
    long d = dst[e];
    atomicAdd(&out[d * 2 + c], feat[s * 2 + c]);
}

// ---------------------------------------------------------------------------
// Fused dual-GEMM layer:  out = act( agg @ Wrel^T + x @ Wroot^T + b )
// WMMA f32 16x16x4. Each wave computes a 16-node x 64-col tile with 4 v8f
// accumulators, 16 K-steps, 2 WMMAs per (kstep,ntile) -> 128 WMMAs/wave.
// Weights are staged once per block into LDS.
//
// A fragment (16x4 f32): lane<16 holds row=lane, K={k0,k0+1};
//                        lane>=16 holds row=lane-16, K={k0+2,k0+3}.
// B fragment (4x16 f32): lane<16 holds col=lane, K={k0,k0+1};
//                        lane>=16 holds col=lane-16, K={k0+2,k0+3}.
// B element [k][n] = W[n][k] (B = W^T), so a lane reads W[col][koff..koff+1]
// which is a contiguous aligned float2 -> single ds_load_b64.
// C/D: vgpr r, lane -> row = r + 8*(lane>>4), col = lane&15.
// ---------------------------------------------------------------------------
#define GEMM_WAVES 8  // 256 threads, 128 nodes per block

__global__ __launch_bounds__(256) void gemm_dual_wmma(
    const float* __restrict__ agg, const float* __restrict__ xin,
    const float* __restrict__ Wrel, const float* __restrict__ Wroot,
    const float* __restrict__ bias, float* __restrict__ out,
    int nNodes, int doRelu) {
    __shared__ float sWrel[64 * 64];
    __shared__ float sWroot[64 * 64];
    __shared__ float sBias[64];

    int tid = threadIdx.x;
    for (int i = tid; i < 64 * 64; i += blockDim.x) {
        sWrel[i]  = Wrel[i];
        sWroot[i] = Wroot[i];
    }
    if (tid < 64) sBias[tid] = bias[tid];
    __syncthreads();

    const int wave = tid >> 5;
    const int lane = tid & 31;
    const int half = lane >> 4;   // 0: K pair {0,1}, 1: K pair {2,3}
    const int l15  = lane & 15;

    const long tile  = (long)blockIdx.x * GEMM_WAVES + wave;
    const long mbase = tile * 16;
    if (mbase >= (long)nNodes) return;  // wave-granular guard: EXEC all-ones

    v8f c[4] = {v8f{}, v8f{}, v8f{}, v8f{}};

    const float* aggRow = agg + (mbase + l15) * 64;
    const float* xRow   = xin + (mbase + l15) * 64;

#pragma unroll
    for (int k0 = 0; k0 < 64; k0 += 4) {
        const int koff = k0 + 2 * half;
        v2f aAgg = *(const v2f*)(aggRow + koff);
        v2f aX   = *(const v2f*)(xRow + koff);
#pragma unroll
        for (int nt = 0; nt < 4; ++nt) {
            const int col = nt * 16 + l15;
            v2f bRel  = *(const v2f*)(&sWrel[col * 64 + koff]);
            v2f bRoot = *(const v2f*)(&sWroot[col * 64 + koff]);
            c[nt] = __builtin_amdgcn_wmma_f32_16x16x4_f32(
                false, aAgg, false, bRel, (short)0, c[nt], false, false);
            c[nt] = __builtin_amdgcn_wmma_f32_16x16x4_f32(
                false, aX, false, bRoot, (short)0, c[nt], false, false);
        }
    }

#pragma unroll
    for (int nt = 0; nt < 4; ++nt) {
        const int col = nt * 16 + l15;
        const float b = sBias[col];
#pragma unroll
        for (int r = 0; r < 8; ++r) {
            const int row = r + 8 * half;
            float v = c[nt][r] + b;
            if (doRelu) v = fmaxf(v, 0.0f);
            out[(mbase + row) * 64 + col] = v;
        }
    }
}

// ---------------------------------------------------------------------------
// Layer 3 node-local part: t = h @ Wrel3^T (pre-projection for the scatter)
// and out = h @ Wroot3^T + b3 (initializes d_out so scatter_add2 lands on it).
// OUT_DIM = 2, so plain VALU dot products are the right tool here.
// ---------------------------------------------------------------------------
__global__ void layer3_node(const float* __restrict__ h,
                            const float* __restrict__ Wrel3,
                            const float* __restrict__ Wroot3,
                            const float* __restrict__ b3,
                            float* __restrict__ t, float* __restrict__ out,
                            int nNodes) {
    int i = blockIdx.x * blockDim.x + threadIdx.x;
    if (i >= nNodes) return;
    const float* row = h + (long)i * 64;
    float t0 = 0.f, t1 = 0.f, o0 = 0.f, o1 = 0.f;
#pragma unroll 8
    for (int k = 0; k < 64; ++k) {
        const float v = row[k];
        t0 = fmaf(v, Wrel3[k], t0);
        t1 = fmaf(v, Wrel3[64 + k], t1);
        o0 = fmaf(v, Wroot3[k], o0);
        o1 = fmaf(v, Wroot3[64 + k], o1);
    }
    t[(long)i * 2 + 0]   = t0;
    t[(long)i * 2 + 1]   = t1;
    out[(long)i * 2 + 0] = o0 + b3[0];
    out[(long)i * 2 + 1] = o1 + b3[1];
}

// ---------------------------------------------------------------------------
extern "C" void kernel_launch(void* const* d_in, const int* in_sizes, int n_in,
                              void* d_out, int out_size, void* d_ws, size_t ws_size,
                              hipStream_t stream) {
    const float* x      = (const float*)d_in[0];
    const int*   eidx   = (const int*)d_in[1];
    const float* W_rel1 = (const float*)d_in[2];
    const float* W_root1= (const float*)d_in[3];
    const float* b1     = (const float*)d_in[4];
    const float* W_rel2 = (const float*)d_in[5];
    const float* W_root2= (const float*)d_in[6];
    const float* b2     = (const float*)d_in[7];
    const float* W_rel3 = (const float*)d_in[8];
    const float* W_root3= (const float*)d_in[9];
    const float* b3     = (const float*)d_in[10];
    float* out = (float*)d_out;

    const int  nNodes = in_sizes[0] / 64;
    const long nEdges = (long)in_sizes[1] / 2;
    const int* src = eidx;
    const int* dst = eidx + nEdges;

    // Workspace layout (floats): agg | h1 | h2 ; t aliases h1 (dead by then)
    const long nodeFeat = (long)nNodes * 64;
    float* agg = (float*)d_ws;
    float* h1  = agg + nodeFeat;
    float* h2  = h1 + nodeFeat;
    float* t   = h1;  // reuse: h1 is dead after layer-2 GEMM

    const int  ZB = 256;
    const long zeroGrid = 2048;
    const long sc64Work = nEdges * 64;
    const long sc64Grid = (sc64Work + 255) / 256;
    const long sc2Work  = nEdges * 2;
    const long sc2Grid  = (sc2Work + 255) / 256;
    const int  gemmGrid = (int)(((long)nNodes + 16L * GEMM_WAVES - 1) / (16L * GEMM_WAVES));
    const int  nodeGrid = (nNodes + 255) / 256;

    // ---- Layer 1 ----
    zero_kernel<<<zeroGrid, ZB, 0, stream>>>(agg, nodeFeat);
    scatter_add64<<<sc64Grid, 256, 0, stream>>>(x, src, dst, agg, sc64Work);
    gemm_dual_wmma<<<gemmGrid, 256, 0, stream>>>(agg, x, W_rel1, W_root1, b1,
                                                 h1, nNodes, 1);
    // ---- Layer 2 ----
    zero_kernel<<<zeroGrid, ZB, 0, stream>>>(agg, nodeFeat);
    scatter_add64<<<sc64Grid, 256, 0, stream>>>(h1, src, dst, agg, sc64Work);
    gemm_dual_wmma<<<gemmGrid, 256, 0, stream>>>(agg, h1, W_rel2, W_root2, b2,
                                                 h2, nNodes, 1);
    // ---- Layer 3 (pre-project to 2 dims, then 2-wide scatter) ----
    layer3_node<<<nodeGrid, 256, 0, stream>>>(h2, W_rel3, W_root3, b3, t, out,
                                              nNodes);
    scatter_add2<<<sc2Grid, 256, 0, stream>>>(t, src, dst, out, sc2Work);
}